// ChamferDistanceLoss_25563645346662
// MI455X (gfx1250) — compile-verified
//
#include <hip/hip_runtime.h>

typedef float v2f __attribute__((ext_vector_type(2)));
typedef float v8f __attribute__((ext_vector_type(8)));

__global__ void chamfer_zero_out(float* out) { out[0] = 0.0f; }

// One wave computes 16 rows of P against all Nq columns of Q.
// dist[m,n] = |p|^2 + |q|^2 - 2 p.q  is computed as one WMMA per 16x16 tile:
//   A[m,:] = (-2px, -2py, -2pz, 1)   (K padded from 3 to 4)
//   B[:,n] = ( qx,   qy,   qz, |q|^2)
//   D[m,n] = -2 p.q + |q|^2          (+ |p|^2 added after the row-min)
__global__ __launch_bounds__(256) void chamfer_dir_wmma(
    const float* __restrict__ Pbase, const float* __restrict__ Qbase,
    int Np, int Nq, float scale, float* __restrict__ out)
{
    const int lane  = threadIdx.x & 31;
    const int wave  = threadIdx.x >> 5;
    const int batch = blockIdx.y;
    const int row0  = (blockIdx.x * 8 + wave) * 16;
    if (row0 >= Np) return;  // wave-uniform: EXEC stays all-ones

    const float* __restrict__ P = Pbase + (size_t)batch * Np * 3;
    const float* __restrict__ Q = Qbase + (size_t)batch * Nq * 3;

    const int  sub = lane & 15;
    const bool hi  = lane >= 16;

    // Each lane loads the full point of its row (both halves need it).
    const int m  = row0 + sub;
    const float px = P[3 * m + 0];
    const float py = P[3 * m + 1];
    const float pz = P[3 * m + 2];
    const float p2 = px * px + py * py + pz * pz;

    // A-matrix 16x4 f32 layout: lanes 0-15 -> (K0,K1), lanes 16-31 -> (K2,K3)
    v2f a;
    a.x = hi ? (-2.0f * pz) : (-2.0f * px);
    a.y = hi ? 1.0f         : (-2.0f * py);

    v8f minv;
#pragma unroll
    for (int i = 0; i < 8; ++i) minv[i] = 3.0e38f;

#pragma unroll 2
    for (int ct = 0; ct < Nq; ct += 16) {
        const int n = ct + sub;
        const float qx = Q[3 * n + 0];
        const float qy = Q[3 * n + 1];
        const float qz = Q[3 * n + 2];
        const float q2 = qx * qx + qy * qy + qz * qz;

        // B-matrix 4x16 f32 layout: lanes 0-15 -> (K0,K1), lanes 16-31 -> (K2,K3)
        v2f b;
        b.x = hi ? qz : qx;
        b.y = hi ? q2 : qy;

        v8f c = {};
        // D = A x B + 0  -> one v_wmma_f32_16x16x4_f32 per 16x16 tile
        c = __builtin_amdgcn_wmma_f32_16x16x4_f32(
                /*neg_a=*/false, a, /*neg_b=*/false, b,
                /*c_mod=*/(short)0, c, /*reuse_a=*/false, /*reuse_b=*/false);

#pragma unroll
        for (int i = 0; i < 8; ++i) minv[i] = fminf(minv[i], c[i]);
    }

    // Row-min: D VGPR j holds row (row0+j) across lanes 0-15 and row (row0+8+j)
    // across lanes 16-31. 4 xor-shuffle steps reduce within each 16-lane half.
    float summin = 0.0f;
#pragma unroll
    for (int i = 0; i < 8; ++i) {
        float r = minv[i];
        r = fminf(r, __shfl_xor(r, 1, 32));
        r = fminf(r, __shfl_xor(r, 2, 32));
        r = fminf(r, __shfl_xor(r, 4, 32));
        r = fminf(r, __shfl_xor(r, 8, 32));
        summin += r;  // identical in all lanes of each half
    }

    // Count each half's row-min sum once (lanes 0 and 16); add |p|^2 once per
    // row (lanes 0-15). Then a full wave32 sum-reduce -> one atomic per wave.
    float partial = (sub == 0 ? summin : 0.0f) + (hi ? 0.0f : p2);
#pragma unroll
    for (int off = 16; off >= 1; off >>= 1)
        partial += __shfl_xor(partial, off, 32);

    if (lane == 0) atomicAdd(out, partial * scale);
}

extern "C" void kernel_launch(void* const* d_in, const int* in_sizes, int n_in,
                              void* d_out, int out_size, void* d_ws, size_t ws_size,
                              hipStream_t stream) {
    const float* p1 = (const float*)d_in[0];
    const float* p2 = (const float*)d_in[1];
    float* out = (float*)d_out;

    const int N = 4096, M = 4096, D = 3;
    const int B = in_sizes[0] / (N * D);            // 16
    const float scale = 1.0f / (float)B;            // CD_W1 = CD_W2 = 1

    chamfer_zero_out<<<1, 1, 0, stream>>>(out);

    dim3 blk(256);
    dim3 g1((N / 16 + 7) / 8, B);
    chamfer_dir_wmma<<<g1, blk, 0, stream>>>(p1, p2, N, M, scale, out);
    dim3 g2((M / 16 + 7) / 8, B);
    chamfer_dir_wmma<<<g2, blk, 0, stream>>>(p2, p1, M, N, scale, out);
}